// DARUAN_45311904972849
// MI455X (gfx1250) — compile-verified
//
#include <hip/hip_runtime.h>
#include <math.h>

// Workload: per-(batch,dim) single-qubit re-uploading circuit -> <Z>.
// Purely elementwise (no matmul), so WMMA does not apply. CDNA5-specific
// paths used instead: global_load_async_to_lds_b32 (ASYNCcnt) to stage
// per-dim gate tables into LDS, s_wait_asynccnt, global_prefetch_b8.
//
// Math: Bloch-vector formulation. v=(1,0,0) after H.
//   RZ(t): (x,y) -> (x cos t - y sin t, x sin t + y cos t)
//   RY(t): (x,z) -> (x cos t + z sin t, -x sin t + z cos t)
// Fuse RZ(enc_r) with RZ(theta0[r+1]) (both z-rotations commute):
//   init: apply RZ(theta0[0]) to (1,0,0) -> (c00, s00, 0)   [per-dim const]
//   loop r=0..R-1: RY(theta1[r]); RZ( fma(w_r, x, bias_r + theta0[r+1]) )
//   final: RY(theta1[R]); out = z*pw + pb
//
// Per-dim table (field-major, stride=dim in ws / TD in LDS), NF = 4R+6:
//   f[0..R]        cy[r]  = cos(theta1[r])
//   f[R+1..2R+1]   sy[r]  = sin(theta1[r])
//   f[2R+2..3R+1]  w[r]   = preacts_weight[d,r]
//   f[3R+2..4R+1]  b2[r]  = preacts_bias[d,r] + theta0[r+1]
//   f[4R+2]=c00  f[4R+3]=s00  f[4R+4]=pw  f[4R+5]=pb
//
// The batch loop processes 2 rows per iteration: two independent Bloch
// chains expose ILP so the wave32 SIMD can dual-issue (VOPD) and hide
// v_sin/v_cos latency of one chain under ALU work of the other.

#define TD 256   // dims per block (== blockDim.x)
#define TB 16    // batch rows per block

__global__ void qk_precompute(const float* __restrict__ theta,
                              const float* __restrict__ paw,
                              const float* __restrict__ pab,
                              const float* __restrict__ postw,
                              const float* __restrict__ postb,
                              float* __restrict__ tab, int dim, int reps) {
    int d = blockIdx.x * blockDim.x + threadIdx.x;
    if (d >= dim) return;
    const float* th = theta + (size_t)d * (size_t)(reps + 1) * 2;
    for (int r = 0; r <= reps; ++r) {
        float s, c;
        __sincosf(th[2 * r + 1], &s, &c);
        tab[(size_t)r * dim + d] = c;                       // cy[r]
        tab[(size_t)(reps + 1 + r) * dim + d] = s;          // sy[r]
    }
    int bw = 2 * (reps + 1);
    for (int r = 0; r < reps; ++r) {
        tab[(size_t)(bw + r) * dim + d] = paw[(size_t)d * reps + r];
        tab[(size_t)(bw + reps + r) * dim + d] =
            pab[(size_t)d * reps + r] + th[2 * (r + 1)];    // fused bias
    }
    int bi = bw + 2 * reps;
    float s0, c0;
    __sincosf(th[0], &s0, &c0);
    tab[(size_t)(bi + 0) * dim + d] = c0;
    tab[(size_t)(bi + 1) * dim + d] = s0;
    tab[(size_t)(bi + 2) * dim + d] = postw[d];
    tab[(size_t)(bi + 3) * dim + d] = postb[d];
}

template <int R, bool STAGE>
__global__ __launch_bounds__(TD) void qk_main(
    const float* __restrict__ x, const float* __restrict__ tab,
    const float* __restrict__ theta, const float* __restrict__ paw,
    const float* __restrict__ pab, const float* __restrict__ postw,
    const float* __restrict__ postb, float* __restrict__ out,
    int batch, int dim) {
    constexpr int NF = 4 * R + 6;
    __shared__ float smem[NF * TD];
    const int t = threadIdx.x;
    const int d0 = blockIdx.x * TD;
    const int d = d0 + t;
    const int b0 = blockIdx.y * TB;

    if (STAGE) {
        // CDNA5 async copy engine: per-lane global -> LDS, ASYNCcnt-tracked.
        unsigned lds_base = (unsigned)(size_t)(void*)smem; // low 32b = LDS offset
        unsigned voff = (unsigned)(t * 4);
#pragma unroll
        for (int f = 0; f < NF; ++f) {
            const float* gp = tab + (size_t)f * dim + d0;   // uniform -> SGPR pair
            unsigned lo = lds_base + (unsigned)((f * TD + t) * 4);
            asm volatile("global_load_async_to_lds_b32 %0, %1, %2"
                         :: "v"(lo), "v"(voff), "s"(gp) : "memory");
        }
        asm volatile("s_wait_asynccnt 0" ::: "memory");
    } else {
        // Fallback: each thread builds its own dim's table in LDS.
        if (d < dim) {
            const float* th = theta + (size_t)d * (R + 1) * 2;
#pragma unroll
            for (int r = 0; r <= R; ++r) {
                float s, c;
                __sincosf(th[2 * r + 1], &s, &c);
                smem[r * TD + t] = c;
                smem[(R + 1 + r) * TD + t] = s;
            }
#pragma unroll
            for (int r = 0; r < R; ++r) {
                smem[(2 * (R + 1) + r) * TD + t] = paw[(size_t)d * R + r];
                smem[(2 * (R + 1) + R + r) * TD + t] =
                    pab[(size_t)d * R + r] + th[2 * (r + 1)];
            }
            float s0, c0;
            __sincosf(th[0], &s0, &c0);
            int bi = 4 * R + 2;
            smem[(bi + 0) * TD + t] = c0;
            smem[(bi + 1) * TD + t] = s0;
            smem[(bi + 2) * TD + t] = postw[d];
            smem[(bi + 3) * TD + t] = postb[d];
        }
    }
    __syncthreads();
    if (d >= dim) return;

    // Hoist tables to registers: inner loop becomes pure VALU + loads/stores.
    float cy[R + 1], sy[R + 1], w[R], b2[R];
#pragma unroll
    for (int r = 0; r <= R; ++r) {
        cy[r] = smem[r * TD + t];
        sy[r] = smem[(R + 1 + r) * TD + t];
    }
#pragma unroll
    for (int r = 0; r < R; ++r) {
        w[r] = smem[(2 * (R + 1) + r) * TD + t];
        b2[r] = smem[(2 * (R + 1) + R + r) * TD + t];
    }
    const int bi = 4 * R + 2;
    const float bx0 = smem[(bi + 0) * TD + t];
    const float by0 = smem[(bi + 1) * TD + t];
    const float pw = smem[(bi + 2) * TD + t];
    const float pb = smem[(bi + 3) * TD + t];

    int bmax = b0 + TB;
    if (bmax > batch) bmax = batch;

    int b = b0;
    // 2 rows per iteration: two independent chains for VOPD dual-issue.
    for (; b + 1 < bmax; b += 2) {
        float xv0 = x[(size_t)b * dim + d];
        float xv1 = x[(size_t)(b + 1) * dim + d];
        if (b + 2 < bmax)
            __builtin_prefetch(&x[(size_t)(b + 2) * dim + d], 0, 3);
        float ax = bx0, ay = by0, az = 0.0f;
        float ux = bx0, uy = by0, uz = 0.0f;
#pragma unroll
        for (int r = 0; r < R; ++r) {
            // RY(theta1[r]) on both chains
            float n0 = cy[r] * ax + sy[r] * az;
            az = cy[r] * az - sy[r] * ax;
            ax = n0;
            float n1 = cy[r] * ux + sy[r] * uz;
            uz = cy[r] * uz - sy[r] * ux;
            ux = n1;
            // RZ(enc + theta0[r+1]) on both chains
            float s0, c0, s1, c1;
            __sincosf(fmaf(w[r], xv0, b2[r]), &s0, &c0);
            __sincosf(fmaf(w[r], xv1, b2[r]), &s1, &c1);
            n0 = c0 * ax - s0 * ay;
            ay = fmaf(s0, ax, c0 * ay);
            ax = n0;
            n1 = c1 * ux - s1 * uy;
            uy = fmaf(s1, ux, c1 * uy);
            ux = n1;
        }
        float z0 = cy[R] * az - sy[R] * ax;
        float z1 = cy[R] * uz - sy[R] * ux;
        out[(size_t)b * dim + d] = fmaf(z0, pw, pb);
        out[(size_t)(b + 1) * dim + d] = fmaf(z1, pw, pb);
    }
    // remainder row (batch odd within tile)
    for (; b < bmax; ++b) {
        float xv = x[(size_t)b * dim + d];
        float ax = bx0, ay = by0, az = 0.0f;
#pragma unroll
        for (int r = 0; r < R; ++r) {
            float nx = cy[r] * ax + sy[r] * az;
            az = cy[r] * az - sy[r] * ax;
            ax = nx;
            float s, c;
            __sincosf(fmaf(w[r], xv, b2[r]), &s, &c);
            nx = c * ax - s * ay;
            ay = fmaf(s, ax, c * ay);
            ax = nx;
        }
        float z = cy[R] * az - sy[R] * ax;
        out[(size_t)b * dim + d] = fmaf(z, pw, pb);
    }
}

// Safety net for unexpected shapes (reps != 8 or dim % TD != 0).
__global__ void qk_generic(const float* __restrict__ x,
                           const float* __restrict__ theta,
                           const float* __restrict__ paw,
                           const float* __restrict__ pab,
                           const float* __restrict__ postw,
                           const float* __restrict__ postb,
                           float* __restrict__ out,
                           long long n, int dim, int reps) {
    long long i = (long long)blockIdx.x * blockDim.x + threadIdx.x;
    if (i >= n) return;
    int d = (int)(i % dim);
    float xv = x[i];
    const float* th = theta + (size_t)d * (size_t)(reps + 1) * 2;
    float s, c;
    __sincosf(th[0], &s, &c);
    float bx = c, by = s, bz = 0.0f;
    for (int r = 0; r < reps; ++r) {
        float cyv, syv;
        __sincosf(th[2 * r + 1], &syv, &cyv);
        float nx = cyv * bx + syv * bz;
        bz = cyv * bz - syv * bx;
        bx = nx;
        float ang = fmaf(paw[(size_t)d * reps + r], xv,
                         pab[(size_t)d * reps + r] + th[2 * (r + 1)]);
        __sincosf(ang, &s, &c);
        nx = c * bx - s * by;
        by = fmaf(s, bx, c * by);
        bx = nx;
    }
    float cyv, syv;
    __sincosf(th[2 * reps + 1], &syv, &cyv);
    float z = cyv * bz - syv * bx;
    out[i] = fmaf(z, postw[d], postb[d]);
}

extern "C" void kernel_launch(void* const* d_in, const int* in_sizes, int n_in,
                              void* d_out, int out_size, void* d_ws, size_t ws_size,
                              hipStream_t stream) {
    const float* x = (const float*)d_in[0];
    const float* theta = (const float*)d_in[1];
    const float* paw = (const float*)d_in[2];
    const float* pab = (const float*)d_in[3];
    const float* postw = (const float*)d_in[4];
    const float* postb = (const float*)d_in[5];
    float* out = (float*)d_out;

    int dim = in_sizes[4];
    int batch = (dim > 0) ? in_sizes[0] / dim : 0;
    int reps = (dim > 0) ? in_sizes[2] / dim : 0;

    if (reps == 8 && dim % TD == 0) {
        constexpr int NF = 4 * 8 + 6;
        size_t need = (size_t)NF * (size_t)dim * sizeof(float);
        dim3 grid(dim / TD, (batch + TB - 1) / TB);
        if (ws_size >= need) {
            float* tab = (float*)d_ws;
            qk_precompute<<<(dim + 255) / 256, 256, 0, stream>>>(
                theta, paw, pab, postw, postb, tab, dim, reps);
            qk_main<8, true><<<grid, TD, 0, stream>>>(
                x, tab, theta, paw, pab, postw, postb, out, batch, dim);
        } else {
            qk_main<8, false><<<grid, TD, 0, stream>>>(
                x, nullptr, theta, paw, pab, postw, postb, out, batch, dim);
        }
    } else {
        long long n = (long long)batch * dim;
        if (n > 0) {
            int th = 256;
            long long nb = (n + th - 1) / th;
            qk_generic<<<(unsigned)nb, th, 0, stream>>>(
                x, theta, paw, pab, postw, postb, out, n, dim, reps);
        }
    }
}